// Net_13520557048112
// MI455X (gfx1250) — compile-verified
//
#include <hip/hip_runtime.h>
#include <hip/hip_bf16.h>

typedef __attribute__((ext_vector_type(2))) float v2f;
typedef __attribute__((ext_vector_type(8))) float v8f;

#define LWAVES 4              // waves per block in the WMMA MLP kernel (wave32)
#define TPB    (LWAVES * 32)  // 128 threads

__device__ __forceinline__ void atomAddF(float* p, float v) {
  // hardware global_atomic_add_f32 (no CAS loop)
  unsafeAtomicAdd(p, v);
}

// ---------------------------------------------------------------- zero
__global__ void zero_kernel(float* __restrict__ p, int n) {
  int i = blockIdx.x * blockDim.x + threadIdx.x;
  if (i < n) p[i] = 0.f;
}

// ------------------------------------- feat/agg init (+optional BN+ReLU)
// v = (bnp ? relu(src*scale + shift) : src);  feat = v;  agg = v (agg seeds h = x + sum)
__global__ void init_feat_kernel(const float* __restrict__ src,
                                 const float* __restrict__ bnp,  // [scale[64] | shift[64]] or null
                                 float* __restrict__ feat, float* __restrict__ agg,
                                 long long n4) {
  long long gid = (long long)blockIdx.x * blockDim.x + threadIdx.x;
  if (gid >= n4) return;
  float4 v = ((const float4*)src)[gid];
  if (bnp) {
    int c = (int)(gid & 15) * 4;   // feature column of first element of this float4
    v.x = fmaxf(fmaf(v.x, bnp[c + 0], bnp[64 + c + 0]), 0.f);
    v.y = fmaxf(fmaf(v.y, bnp[c + 1], bnp[64 + c + 1]), 0.f);
    v.z = fmaxf(fmaf(v.z, bnp[c + 2], bnp[64 + c + 2]), 0.f);
    v.w = fmaxf(fmaf(v.w, bnp[c + 3], bnp[64 + c + 3]), 0.f);
  }
  ((float4*)feat)[gid] = v;
  ((float4*)agg)[gid]  = v;
}

// ------------------------------------------------- edge scatter-add
// agg[dst] += feat[src]; one thread handles 4 features of one edge
__global__ void edge_scatter_kernel(const float* __restrict__ feat,
                                    float* __restrict__ agg,
                                    const long long* __restrict__ ei,  // [2,E] int64
                                    long long E) {
  long long gid = (long long)blockIdx.x * blockDim.x + threadIdx.x;
  if (gid >= E * 16) return;
  long long e = gid >> 4;
  int q = (int)(gid & 15) * 4;
  long long s = ei[e];
  long long d = ei[E + e];
  float4 v = *(const float4*)(feat + s * 64 + q);
  float* a = agg + d * 64 + q;
  atomAddF(a + 0, v.x);
  atomAddF(a + 1, v.y);
  atomAddF(a + 2, v.z);
  atomAddF(a + 3, v.w);
}

// ------------------------------------------------- fused 2-layer MLP (WMMA fp32)
// out = relu(relu(in @ W1 + b1) @ W2 + b2); also accumulates BN stats (sum, sumsq).
// One wave handles 16 nodes (M=16); output width 64 = 4 N-tiles of 16; K stepped by 4.
// Weights staged in LDS *pair-swizzled*: sW[kpair][col][2] = {W[2kp][c], W[2kp+1][c]}
// so every B fragment is one 8-byte-aligned ds_load_b64 straight into an even VGPR pair.
__global__ __launch_bounds__(TPB) void mlp_wmma_kernel(
    const float* __restrict__ in,                                  // [N,64]
    const float* __restrict__ W1g, const float* __restrict__ b1g,  // [64,64],[64]
    const float* __restrict__ W2g, const float* __restrict__ b2g,  // [64,64],[64]
    float* __restrict__ out,                                       // [N,64] pre-BN
    float* __restrict__ stats,                                     // [128] sum|sumsq
    int N) {
  __shared__ float sW1[64 * 64];         // swizzled: [(k>>1)*64 + c]*2 + (k&1)
  __shared__ float sW2[64 * 64];
  __shared__ float sB1[64];
  __shared__ float sB2[64];
  __shared__ float sSum[64];
  __shared__ float sSq[64];
  __shared__ float sH[LWAVES][16 * 64];  // per-wave h1 staging (A-fragment relayout)

  const int tid = threadIdx.x;
  for (int i = tid; i < 64 * 64; i += TPB) {
    int k = i >> 6, c = i & 63;
    int j = (((k >> 1) * 64 + c) << 1) + (k & 1);
    sW1[j] = W1g[i];
    sW2[j] = W2g[i];
  }
  if (tid < 64) { sB1[tid] = b1g[tid]; sB2[tid] = b2g[tid]; sSum[tid] = 0.f; sSq[tid] = 0.f; }
  __syncthreads();

  const int wave = tid >> 5;
  const int lane = tid & 31;
  const int lo = lane & 15;   // A: M row / B,C,D: N column
  const int hi = lane >> 4;   // K-half selector (A,B); M+8 selector (C,D)
  const int nodeBase = (blockIdx.x * LWAVES + wave) * 16;

  if (nodeBase < N) {  // wave-uniform branch: EXEC stays all-ones for WMMA (N % 16 == 0)
    // ---- GEMM1: h1 = relu(in[16x64] @ W1 + b1) ----
    v8f acc[4] = {};
    const float* rowp = in + (long long)(nodeBase + lo) * 64 + hi * 2;
    #pragma unroll
    for (int kb = 0; kb < 64; kb += 4) {
      v2f a = *(const v2f*)(rowp + kb);                      // global_load_b64
      const float* wp = sW1 + (((kb >> 1) + hi) * 64 + lo) * 2;
      #pragma unroll
      for (int nt = 0; nt < 4; ++nt) {
        v2f b = *(const v2f*)(wp + nt * 32);                 // ds_load_b64, pair-contiguous
        acc[nt] = __builtin_amdgcn_wmma_f32_16x16x4_f32(
            false, a, false, b, (short)0, acc[nt], false, false);
      }
    }
    // bias + ReLU, stage to LDS row-major for A-fragment reload
    float* hbuf = &sH[wave][0];
    #pragma unroll
    for (int nt = 0; nt < 4; ++nt) {
      float bias = sB1[nt * 16 + lo];
      #pragma unroll
      for (int r = 0; r < 8; ++r) {
        float v = fmaxf(acc[nt][r] + bias, 0.f);             // D[m=r+hi*8][n=nt*16+lo]
        hbuf[(r + hi * 8) * 64 + nt * 16 + lo] = v;
      }
    }
    // ---- GEMM2: h2 = relu(h1 @ W2 + b2) ----  (same-wave LDS RAW: DS ops in-order)
    v8f acc2[4] = {};
    const float* hrow = hbuf + lo * 64 + hi * 2;
    #pragma unroll
    for (int kb = 0; kb < 64; kb += 4) {
      v2f a = *(const v2f*)(hrow + kb);                      // ds_load_b64
      const float* wp = sW2 + (((kb >> 1) + hi) * 64 + lo) * 2;
      #pragma unroll
      for (int nt = 0; nt < 4; ++nt) {
        v2f b = *(const v2f*)(wp + nt * 32);
        acc2[nt] = __builtin_amdgcn_wmma_f32_16x16x4_f32(
            false, a, false, b, (short)0, acc2[nt], false, false);
      }
    }
    // bias + ReLU + global store + BN stats (per-lane partials -> LDS atomics)
    #pragma unroll
    for (int nt = 0; nt < 4; ++nt) {
      float bias = sB2[nt * 16 + lo];
      float cs = 0.f, cq = 0.f;
      #pragma unroll
      for (int r = 0; r < 8; ++r) {
        float v = fmaxf(acc2[nt][r] + bias, 0.f);
        out[(long long)(nodeBase + r + hi * 8) * 64 + nt * 16 + lo] = v;
        cs += v;
        cq += v * v;
      }
      atomicAdd(&sSum[nt * 16 + lo], cs);  // ds_add_f32
      atomicAdd(&sSq[nt * 16 + lo], cq);
    }
  }
  __syncthreads();
  if (tid < 64) {
    atomAddF(&stats[tid], sSum[tid]);
    atomAddF(&stats[64 + tid], sSq[tid]);
  }
}

// ---------------------------------------------- BN finalize -> (scale, shift)
__global__ void bn_finalize_kernel(const float* __restrict__ stats,
                                   const float* __restrict__ gamma,
                                   const float* __restrict__ beta,
                                   float* __restrict__ bnp, float invN) {
  int c = threadIdx.x;  // 64 threads
  float mu  = stats[c] * invN;
  float var = stats[64 + c] * invN - mu * mu;   // biased variance
  float rstd = rsqrtf(var + 1e-5f);
  float sc = gamma[c] * rstd;
  bnp[c] = sc;
  bnp[64 + c] = beta[c] - mu * sc;
}

// ---------------------------------------------- graph pooling (BN+ReLU fused)
__global__ void pool_kernel(const float* __restrict__ h,
                            const float* __restrict__ bnp,
                            const long long* __restrict__ batch,
                            float* __restrict__ pooled, long long n16) {
  long long gid = (long long)blockIdx.x * blockDim.x + threadIdx.x;
  if (gid >= n16) return;
  long long node = gid >> 4;
  int c = (int)(gid & 15) * 4;
  long long g = batch[node];
  float4 v = *(const float4*)(h + node * 64 + c);
  float* p = pooled + g * 64 + c;
  atomAddF(p + 0, fmaxf(fmaf(v.x, bnp[c + 0], bnp[64 + c + 0]), 0.f));
  atomAddF(p + 1, fmaxf(fmaf(v.y, bnp[c + 1], bnp[64 + c + 1]), 0.f));
  atomAddF(p + 2, fmaxf(fmaf(v.z, bnp[c + 2], bnp[64 + c + 2]), 0.f));
  atomAddF(p + 3, fmaxf(fmaf(v.w, bnp[c + 3], bnp[64 + c + 3]), 0.f));
}

// ---------------------------------------------- final FC [G,64] @ [64,1] + b
__global__ void fc_kernel(const float* __restrict__ pooled,
                          const float* __restrict__ w, const float* __restrict__ b,
                          float* __restrict__ outp, int G) {
  int g = blockIdx.x * blockDim.x + threadIdx.x;
  if (g >= G) return;
  float s = b[0];
  #pragma unroll 8
  for (int c = 0; c < 64; ++c) s = fmaf(pooled[g * 64 + c], w[c], s);
  outp[g] = s;
}

extern "C" void kernel_launch(void* const* d_in, const int* in_sizes, int n_in,
                              void* d_out, int out_size, void* d_ws, size_t ws_size,
                              hipStream_t stream) {
  (void)n_in; (void)ws_size;
  // Flat dict order: x, edge_index, batch, {l1,l2,l3:{w1,b1,w2,b2,gamma,beta}}, fc_w, fc_b
  const float* x          = (const float*)d_in[0];
  const long long* ei     = (const long long*)d_in[1];  // int64 [2,E]
  const long long* batch  = (const long long*)d_in[2];  // int64 [N]
  const float *w1[3], *b1[3], *w2[3], *b2[3], *gm[3], *bt[3];
  for (int i = 0; i < 3; ++i) {
    int base = 3 + 6 * i;
    w1[i] = (const float*)d_in[base + 0];
    b1[i] = (const float*)d_in[base + 1];
    w2[i] = (const float*)d_in[base + 2];
    b2[i] = (const float*)d_in[base + 3];
    gm[i] = (const float*)d_in[base + 4];
    bt[i] = (const float*)d_in[base + 5];
  }
  const float* fcw = (const float*)d_in[21];
  const float* fcb = (const float*)d_in[22];

  const int N = in_sizes[0] / 64;          // 100000 (multiple of 16)
  const long long E = in_sizes[1] / 2;     // 800000
  const int G = out_size;                  // 512

  const long long NF = (long long)N * 64;
  float* feat   = (float*)d_ws;
  float* agg    = feat + NF;
  float* hpre   = agg + NF;
  float* stats  = hpre + NF;
  float* bnp    = stats + 128;
  float* pooled = bnp + 128;

  const long long n4 = NF / 4;
  const int blocks4 = (int)((n4 + 255) / 256);
  const int blocksE = (int)((E * 16 + 255) / 256);
  const int blocksM = (N + LWAVES * 16 - 1) / (LWAVES * 16);
  const long long n16 = (long long)N * 16;

  const float* layerIn = x;
  for (int l = 0; l < 3; ++l) {
    init_feat_kernel<<<blocks4, 256, 0, stream>>>(layerIn, l == 0 ? nullptr : bnp,
                                                  feat, agg, n4);
    edge_scatter_kernel<<<blocksE, 256, 0, stream>>>(feat, agg, ei, E);
    zero_kernel<<<1, 128, 0, stream>>>(stats, 128);
    mlp_wmma_kernel<<<blocksM, TPB, 0, stream>>>(agg, w1[l], b1[l], w2[l], b2[l],
                                                 hpre, stats, N);
    bn_finalize_kernel<<<1, 64, 0, stream>>>(stats, gm[l], bt[l], bnp, 1.0f / (float)N);
    layerIn = hpre;
  }
  zero_kernel<<<(G * 64 + 255) / 256, 256, 0, stream>>>(pooled, G * 64);
  pool_kernel<<<(int)((n16 + 255) / 256), 256, 0, stream>>>(hpre, bnp, batch, pooled, n16);
  fc_kernel<<<(G + 255) / 256, 256, 0, stream>>>(pooled, fcw, fcb, (float*)d_out, G);
}